// Model_89000312308280
// MI455X (gfx1250) — compile-verified
//
#include <hip/hip_runtime.h>
#include <math.h>

// ---------------------------------------------------------------------------
// Problem constants (from reference: n=128, t=2048, c=32)
// ---------------------------------------------------------------------------
#define N_    128
#define T_    2048
#define C_    32
#define R_    (N_ * C_)          // 4096 rows
#define MASK_ELEMS ((size_t)N_ * T_ * C_)   // 8388608

typedef __attribute__((ext_vector_type(2))) float v2f;
typedef __attribute__((ext_vector_type(8))) float v8f;

// ---------------------------------------------------------------------------
// Wave32-wide exact f32 sum using the matrix pipe:
// A (16x4) holds one value per lane (K1/K3 = 0), B = ones, C = 0.
// D[m][n] = rowsum[m] = x[m] + x[m+16].  Sum of the 8 D VGPRs gives
// lanes 0-15 : sum over M=0..7,  lanes 16-31 : sum over M=8..15.
// Adding the xor-16 partner yields the full wave sum in every lane.
// Requires EXEC == all ones (call uniformly from all 256 threads).
// ---------------------------------------------------------------------------
__device__ __forceinline__ float wave_sum_wmma(float x) {
    v2f a; a.x = x;    a.y = 0.0f;
    v2f b; b.x = 1.0f; b.y = 1.0f;
    v8f c = {};
    v8f d = __builtin_amdgcn_wmma_f32_16x16x4_f32(
        /*neg_a=*/false, a, /*neg_b=*/false, b,
        /*c_mod=*/(short)0, c, /*reuse_a=*/false, /*reuse_b=*/false);
    float s = d[0] + d[1] + d[2] + d[3] + d[4] + d[5] + d[6] + d[7];
    s += __shfl_xor(s, 16);
    return s;
}

// Block-wide (256 threads = 8 waves) sum; `red` is an 8-float LDS scratch.
__device__ __forceinline__ float block_sum(float v, float* red) {
    float w = wave_sum_wmma(v);
    int wave = threadIdx.x >> 5;
    if ((threadIdx.x & 31) == 0) red[wave] = w;
    __syncthreads();
    float s = 0.0f;
#pragma unroll
    for (int i = 0; i < 8; ++i) s += red[i];
    __syncthreads();
    return s;
}

// ---------------------------------------------------------------------------
// Kernel 0: coalesced (n,t,c) -> (n,c,t) transpose so each of the 4096 rows
// becomes a contiguous 8KB blob (avoids 32x gather amplification on a
// 23.3 TB/s part; both tensors fit in the 192MB L2 anyway).
// ---------------------------------------------------------------------------
__global__ void transpose_tc(const float* __restrict__ in, float* __restrict__ out) {
    __shared__ float tile[32][33];
    int n  = blockIdx.y;
    int t0 = blockIdx.x * 32;
    int tx = threadIdx.x;          // 0..31
    int ty = threadIdx.y;          // 0..7
    const float* src = in + ((size_t)n * T_ + t0) * C_;
#pragma unroll
    for (int k = 0; k < 4; ++k) {
        int row = ty + 8 * k;                       // ti within tile
        tile[row][tx] = src[(size_t)row * C_ + tx]; // coalesced along c
    }
    __syncthreads();
    float* dst = out + (size_t)n * C_ * T_ + t0;
#pragma unroll
    for (int k = 0; k < 4; ++k) {
        int cc = ty + 8 * k;                        // channel
        dst[(size_t)cc * T_ + tx] = tile[tx][cc];   // coalesced along t
    }
}

// ---------------------------------------------------------------------------
// Kernel 1: one block (256 threads, 8 waves) per row r = n*32 + c.
//  - LDS bitonic sort (value desc, stable via index tie-break)
//  - serial 2-cluster scan (lane 0)          -> smask
//  - WMMA block reductions for act_score     -> mean_score threshold
//  - refined mask scatter to (n,t,c) output
//  - WMMA block reductions for act/bkg raw scores
// LDS: 4 x 8KB arrays + scratch = ~32KB (WGP has 320KB -> high occupancy).
// ---------------------------------------------------------------------------
__global__ void row_pipeline(const float* __restrict__ sT,
                             const float* __restrict__ cT,
                             float* __restrict__ outMask,
                             float* __restrict__ actRaw,
                             float* __restrict__ bkgRaw) {
    __shared__ float sval[T_];
    __shared__ int   sidx[T_];
    __shared__ float smask[T_];
    __shared__ float ssc[T_];
    __shared__ float red[8];

    const int r   = blockIdx.x;
    const int tid = threadIdx.x;
    const float* srow = sT + (size_t)r * T_;
    const float* crow = cT + (size_t)r * T_;

    // load row + identity index
    for (int j = tid; j < T_; j += 256) { sval[j] = srow[j]; sidx[j] = j; }
    __syncthreads();

    // ---- bitonic sort, descending, index tie-break (matches stable argsort(-s))
    for (int k = 2; k <= T_; k <<= 1) {
        for (int jj = k >> 1; jj > 0; jj >>= 1) {
            for (int i = tid; i < T_; i += 256) {
                int ixj = i ^ jj;
                if (ixj > i) {
                    float v1 = sval[i], v2 = sval[ixj];
                    int   i1 = sidx[i], i2 = sidx[ixj];
                    // "ixj's element should precede i's" in descending order
                    bool before = (v2 > v1) || (v2 == v1 && i2 < i1);
                    bool up     = ((i & k) == 0);
                    if (before == up) {
                        sval[i] = v2; sval[ixj] = v1;
                        sidx[i] = i2; sidx[ixj] = i1;
                    }
                }
            }
            __syncthreads();
        }
    }

    // ---- temporal clustering: strictly sequential recurrence (lane 0)
    if (tid == 0) {
        float ps = sval[0],      pn = 1.0f;
        float ns = sval[T_ - 1], nn = 1.0f;
        smask[0]      = 1.0f;
        smask[T_ - 1] = 0.0f;
        for (int j = 1; j <= T_ - 2; ++j) {
            float v  = sval[j];
            float cf = (fabsf(v - ps / pn) <= fabsf(v - ns / nn)) ? 1.0f : 0.0f;
            smask[j] = cf;
            ps += v * cf;            pn += cf;
            ns += v * (1.0f - cf);   nn += (1.0f - cf);
        }
    }
    __syncthreads();

    // ---- gather cas by sort order (hot 8KB region -> cache-friendly)
    for (int j = tid; j < T_; j += 256) ssc[j] = crow[sidx[j]];
    __syncthreads();

    // ---- mask refining: act_score = sum(sc*rank*m) / max(sum(rank*m),1)
    float pa = 0.0f, pb = 0.0f;
    for (int j = tid; j < T_; j += 256) {
        float rk = 1.0f / (float)(j + 2);
        float rs = rk * smask[j];
        pa += rs;
        pb += ssc[j] * rs;
    }
    float act_num   = block_sum(pa, red);
    float act_wsum  = block_sum(pb, red);
    float act_score = act_wsum / fmaxf(act_num, 1.0f);

    float pc = 0.0f;
    for (int j = tid; j < T_; j += 256)
        pc += (ssc[j] >= act_score) ? 1.0f : 0.0f;
    float cnt = block_sum(pc, red);          // exact: integer-valued f32 sums
    int t_m = (int)(cnt + 0.5f) - 1;
    t_m = t_m < 0 ? 0 : (t_m > T_ - 1 ? T_ - 1 : t_m);
    float mean_score = sval[t_m];

    // ---- refined = m * (s >= mean_score), scattered back to (n,t,c)
    const int nidx = r >> 5, cidx = r & 31;
    float* outBase = outMask + (size_t)nidx * T_ * C_ + cidx;
    for (int j = tid; j < T_; j += 256) {
        float rm = smask[j] * ((sval[j] >= mean_score) ? 1.0f : 0.0f);
        smask[j] = rm;                              // same-thread reuse below
        outBase[(size_t)sidx[j] * C_] = rm;
    }

    // ---- raw act/bkg scores with refined mask
    float qa = 0.0f, qb = 0.0f, qc = 0.0f, qd = 0.0f;
    for (int j = tid; j < T_; j += 256) {
        float rk = 1.0f / (float)(j + 2);
        float rm = smask[j];
        float sc = ssc[j];
        qa += rk * rm;
        qb += sc * rk * rm;
        qc += (1.0f - rm);
        qd += sc * (1.0f - rm);
    }
    float a_num = block_sum(qa, red);
    float a_sum = block_sum(qb, red);
    float b_num = block_sum(qc, red);
    float b_sum = block_sum(qd, red);
    if (tid == 0) {
        actRaw[r] = a_sum / fmaxf(a_num, 1.0f);
        bkgRaw[r] = b_sum / fmaxf(b_num, 1.0f);
    }
}

// ---------------------------------------------------------------------------
// Kernel 2: softmax over c=32 per (n, act/bkg). One wave per row.
// ---------------------------------------------------------------------------
__global__ void softmax32(const float* __restrict__ actRaw,
                          const float* __restrict__ bkgRaw,
                          float* __restrict__ out) {
    int b    = blockIdx.x;         // 0..255
    int n    = b >> 1;
    int sel  = b & 1;
    int lane = threadIdx.x;        // 0..31
    const float* src = sel ? bkgRaw : actRaw;
    float v = src[n * C_ + lane];
    float m = v;
#pragma unroll
    for (int o = 16; o > 0; o >>= 1) m = fmaxf(m, __shfl_xor(m, o));
    float e = expf(v - m);
    float s = e;
#pragma unroll
    for (int o = 16; o > 0; o >>= 1) s += __shfl_xor(s, o);
    out[MASK_ELEMS + (size_t)sel * R_ + n * C_ + lane] = e / s;
}

// ---------------------------------------------------------------------------
// Launcher. Workspace layout (floats):
//   [0, R*T)        transposed seg_score (n,c,t)
//   [R*T, 2*R*T)    transposed cas       (n,c,t)
//   [2*R*T, +R)     raw act scores
//   [.., +R)        raw bkg scores
// Total = 67,141,632 bytes.
// ---------------------------------------------------------------------------
extern "C" void kernel_launch(void* const* d_in, const int* in_sizes, int n_in,
                              void* d_out, int out_size, void* d_ws, size_t ws_size,
                              hipStream_t stream) {
    (void)in_sizes; (void)n_in; (void)out_size; (void)ws_size;
    const float* seg_score = (const float*)d_in[0];
    const float* cas       = (const float*)d_in[1];
    float* out = (float*)d_out;

    float* ws     = (float*)d_ws;
    float* sT     = ws;                                  // R_*T_
    float* cT     = ws + (size_t)R_ * T_;                // R_*T_
    float* actRaw = ws + (size_t)2 * R_ * T_;            // R_
    float* bkgRaw = actRaw + R_;                         // R_

    dim3 tgrid(T_ / 32, N_);
    dim3 tblk(32, 8);
    transpose_tc<<<tgrid, tblk, 0, stream>>>(seg_score, sT);
    transpose_tc<<<tgrid, tblk, 0, stream>>>(cas, cT);

    row_pipeline<<<R_, 256, 0, stream>>>(sT, cT, out, actRaw, bkgRaw);

    softmax32<<<2 * N_, 32, 0, stream>>>(actRaw, bkgRaw, out);
}